// GaussianRender_88905823027747
// MI455X (gfx1250) — compile-verified
//
#include <hip/hip_runtime.h>

// GaussianRender: deterministic stable counting-sort by tile id (16x16 tiles),
// plus per-point conic/radius. 4 small passes:
//   K1 : per-chunk 256-bin LDS histogram -> ws (chunkHist)
//   K2a: per-span parallel in-place scan of chunkHist + span sums (64 blocks)
//   K2b: cross-span + cross-bin scan (tiny), tile_counts/offsets out
//   K3 : wave32 ballot-sliced stable rank + async-LDS feature staging + scatter
//
// Output layout (all float32): sorted_feat[N*7] | tile_counts[nb] |
//                              tile_offsets[nb] | order[N]

#define IMG_SIZE_F 1024.0f
#define CHUNK      1024              // points per chunk (1 block)
#define MAX_BINS   256               // supports num_tile <= 16
#define NGROUPS    (CHUNK / 32)      // 32 wave-sized groups per chunk
#define KPT        (CHUNK / 256)     // items per thread = 4
#define TBITS      9                 // 8 bin bits + 1 invalid-marker bit
#define TINVALID   0x100             // marker: bit 8 set, never a real bin
#define NSPAN      64                // parallel scan spans

// ---- gfx1250 async global->LDS path (ASYNCcnt-tracked DMA) ----------------
#if defined(__has_builtin)
#if __has_builtin(__builtin_amdgcn_global_load_async_to_lds_b128) && \
    __has_builtin(__builtin_amdgcn_global_load_async_to_lds_b32)
#define ASYNC_LDS 1
#endif
#endif
#ifndef ASYNC_LDS
#define ASYNC_LDS 0
#endif

#if ASYNC_LDS
#define AS_GLOBAL __attribute__((address_space(1)))
#define AS_LOCAL  __attribute__((address_space(3)))
typedef int i32x4 __attribute__((ext_vector_type(4)));
__device__ __forceinline__ void async_copy_b128(const void* g, void* l) {
    __builtin_amdgcn_global_load_async_to_lds_b128(
        (AS_GLOBAL i32x4*)g, (AS_LOCAL i32x4*)l, 0, 0);
}
__device__ __forceinline__ void async_copy_b32(const void* g, void* l) {
    __builtin_amdgcn_global_load_async_to_lds_b32(
        (AS_GLOBAL int*)g, (AS_LOCAL int*)l, 0, 0);
}
__device__ __forceinline__ void async_wait0() {
#if __has_builtin(__builtin_amdgcn_s_wait_asynccnt)
    __builtin_amdgcn_s_wait_asynccnt(0);
#else
    asm volatile("s_wait_asynccnt 0x0" ::: "memory");
#endif
}
#endif
// ---------------------------------------------------------------------------

__global__ void __launch_bounds__(256)
gs_hist_kernel(const float2* __restrict__ pos, const int* __restrict__ ntile_p,
               int* __restrict__ chunkHist, int N) {
    __shared__ int hist[MAX_BINS];
    const int tid = threadIdx.x;
    hist[tid] = 0;
    __syncthreads();

    const int   nt     = *ntile_p;
    const float ts     = IMG_SIZE_F / (float)nt;   // exact for power-of-two nt
    const float inv_ts = 1.0f / ts;                // exact for power-of-two ts
    const int base = blockIdx.x * CHUNK;

#pragma unroll
    for (int k = 0; k < KPT; ++k) {
        int i = base + k * 256 + tid;
        if (i < N) {
            float2 p = pos[i];
            int tx = (int)floorf(p.x * inv_ts); tx = min(max(tx, 0), nt - 1);
            int ty = (int)floorf(p.y * inv_ts); ty = min(max(ty, 0), nt - 1);
            atomicAdd(&hist[ty * nt + tx], 1);
        }
    }
    __syncthreads();
    chunkHist[blockIdx.x * MAX_BINS + tid] = hist[tid];
}

// K2a: span s converts chunkHist[b][t] to within-span exclusive prefixes and
// writes the span's per-bin total. 64 blocks run in parallel across WGPs.
__global__ void __launch_bounds__(256)
gs_span_scan_kernel(int* __restrict__ chunkHist, int* __restrict__ spanSum,
                    int NB, int CPS) {
    const int t  = threadIdx.x;
    const int s  = blockIdx.x;
    const int b0 = s * CPS;
    const int b1 = min(b0 + CPS, NB);
    int run = 0;
    for (int b = b0; b < b1; ++b) {
        size_t idx = (size_t)b * MAX_BINS + t;
        int v = chunkHist[idx];
        chunkHist[idx] = run;
        run += v;
    }
    spanSum[s * MAX_BINS + t] = run;
}

// K2b: tiny. Scan spanSum across 64 spans per bin, exclusive-scan bins,
// emit tile_counts/tile_offsets, fold bin base into spanSum (global bases).
__global__ void __launch_bounds__(256)
gs_bin_scan_kernel(int* __restrict__ spanSum, const int* __restrict__ ntile_p,
                   float* __restrict__ out, int N) {
    __shared__ int totals[MAX_BINS];
    __shared__ int excl[MAX_BINS];
    const int t = threadIdx.x;

    int run = 0;
#pragma unroll
    for (int s = 0; s < NSPAN; ++s) {
        int idx = s * MAX_BINS + t;
        int v = spanSum[idx];
        spanSum[idx] = run;          // within-bin exclusive prefix over spans
        run += v;
    }
    totals[t] = run;
    __syncthreads();

    if (t == 0) {
        int acc = 0;
        for (int i = 0; i < MAX_BINS; ++i) { excl[i] = acc; acc += totals[i]; }
    }
    __syncthreads();

    const int nt    = *ntile_p;
    const int nbins = nt * nt;
    const size_t countsBase = (size_t)N * 7;
    if (t < nbins) {
        out[countsBase + t]         = (float)totals[t];   // tile_counts
        out[countsBase + nbins + t] = (float)excl[t];     // tile_offsets
    }

    const int base = excl[t];
#pragma unroll
    for (int s = 0; s < NSPAN; ++s) spanSum[s * MAX_BINS + t] += base;
}

__global__ void __launch_bounds__(256)
gs_scatter_kernel(const float2* __restrict__ pos, const float4* __restrict__ cov,
                  const float* __restrict__ opac, const int* __restrict__ ntile_p,
                  const int* __restrict__ chunkOff, const int* __restrict__ spanSum,
                  float* __restrict__ out, int N, int CPS) {
    __shared__ int ghist[NGROUPS * MAX_BINS];   // 32 KB: per-group bin counts
    __shared__ int coff[MAX_BINS];              // combined global bin bases
    __shared__ int coffA[MAX_BINS];             // within-span chunk prefix
    __shared__ int coffB[MAX_BINS];             // span global base
#if ASYNC_LDS
    __shared__ float4 lds_cov[CHUNK];           // 16 KB staged covariances
    __shared__ float  lds_op[CHUNK];            //  4 KB staged opacities
#endif

    const int tid  = threadIdx.x;
    const int wave = tid >> 5;
    const int lane = tid & 31;

    const int   nt     = *ntile_p;
    const float ts     = IMG_SIZE_F / (float)nt;
    const float inv_ts = 1.0f / ts;
    const int chunkBase = blockIdx.x * CHUNK;
    const int span      = blockIdx.x / CPS;

#if ASYNC_LDS
    // Kick off DMA of everything phase 3 needs; overlaps with rank compute.
    async_copy_b32(&chunkOff[blockIdx.x * MAX_BINS + tid], &coffA[tid]);
    async_copy_b32(&spanSum[span * MAX_BINS + tid], &coffB[tid]);
#pragma unroll
    for (int k = 0; k < KPT; ++k) {
        int i = k * 256 + tid;
        int p = chunkBase + i; if (p >= N) p = N - 1;
        async_copy_b128(&cov[p], &lds_cov[i]);
        async_copy_b32(&opac[p], &lds_op[i]);
    }
#else
    coffA[tid] = chunkOff[blockIdx.x * MAX_BINS + tid];
    coffB[tid] = spanSum[span * MAX_BINS + tid];
#endif

#pragma unroll
    for (int k = 0; k < (NGROUPS * MAX_BINS) / 256; ++k) ghist[k * 256 + tid] = 0;
    __syncthreads();

    // Phase 1: each wave handles KPT consecutive 32-point groups. Equal-tile
    // lane mask via bit-sliced ballots (9 ballots for 9-bit keys, wave32).
    int   tb[KPT];
    int   rk[KPT];
    float qx[KPT];
    float qy[KPT];
#pragma unroll
    for (int k = 0; k < KPT; ++k) {
        const int g = wave * KPT + k;
        const int p = chunkBase + g * 32 + lane;
        const bool valid = (p < N);

        int t;
        float2 q = make_float2(0.0f, 0.0f);
        if (valid) {
            q = pos[p];
            int tx = (int)floorf(q.x * inv_ts); tx = min(max(tx, 0), nt - 1);
            int ty = (int)floorf(q.y * inv_ts); ty = min(max(ty, 0), nt - 1);
            t = ty * nt + tx;
        } else {
            t = TINVALID;
        }
#if !ASYNC_LDS
        const int pc = valid ? p : (N - 1);
        __builtin_prefetch(&cov[pc], 0, 3);
        __builtin_prefetch(&opac[pc], 0, 3);
#endif

        unsigned mask = 0xffffffffu;
#pragma unroll
        for (int b = 0; b < TBITS; ++b) {
            unsigned bb = (unsigned)__ballot((t >> b) & 1);
            mask &= ((t >> b) & 1) ? bb : ~bb;
        }
        const int r = __popc(mask & ((1u << lane) - 1u));
        tb[k] = t;
        rk[k] = r;
        qx[k] = q.x;
        qy[k] = q.y;
        if (valid && r == 0) ghist[g * MAX_BINS + t] = __popc(mask);
    }
    __syncthreads();

    // Phase 2: per-bin exclusive scan across the groups (bin t -> bank t%64;
    // 32 lanes of a wave hit 32 distinct banks).
    {
        int run = 0;
#pragma unroll
        for (int g = 0; g < NGROUPS; ++g) {
            int v = ghist[g * MAX_BINS + tid];
            ghist[g * MAX_BINS + tid] = run;
            run += v;
        }
    }
#if ASYNC_LDS
    async_wait0();                    // THIS wave's DMA done (incl. slot tid)
#endif
    coff[tid] = coffA[tid] + coffB[tid];   // each thread folds its own slot
    __syncthreads();                  // all DMA + scans + coff visible

    // Phase 3: compute features and scatter to the stable destination.
    const size_t orderBase = (size_t)N * 7 + 2u * (size_t)(nt * nt);
#pragma unroll
    for (int k = 0; k < KPT; ++k) {
        const int g = wave * KPT + k;
        const int p = chunkBase + g * 32 + lane;
        if (p >= N) continue;
        const int t    = tb[k];
        const int dest = coff[t] + ghist[g * MAX_BINS + t] + rk[k];

#if ASYNC_LDS
        float4 cv = lds_cov[g * 32 + lane];
        float  op = lds_op[g * 32 + lane];
#else
        float4 cv = cov[p];
        float  op = opac[p];
#endif

        float a = cv.x, b = cv.y, c = cv.z, d = cv.w;
        float trace = a + d;
        float det   = a * d - b * c;
        float t1    = 0.5f * trace;
        float t2    = 0.5f * sqrtf(fmaxf(trace * trace - 4.0f * det, 0.0f));
        float radius = fmaxf(t1 - t2, t1 + t2);
        float inv    = 1.0f / det;

        size_t o = (size_t)dest * 7;
        out[o + 0] = qx[k];
        out[o + 1] = qy[k];
        out[o + 2] = d * inv;
        out[o + 3] = -b * inv;
        out[o + 4] = a * inv;
        out[o + 5] = op;
        out[o + 6] = radius;
        out[orderBase + dest] = (float)p;
    }
}

extern "C" void kernel_launch(void* const* d_in, const int* in_sizes, int n_in,
                              void* d_out, int out_size, void* d_ws, size_t ws_size,
                              hipStream_t stream) {
    const float2* pos   = (const float2*)d_in[0];
    const float4* cov   = (const float4*)d_in[1];
    const float*  opac  = (const float*)d_in[2];
    const int*    ntile = (const int*)d_in[3];
    float* out = (float*)d_out;

    const int N   = in_sizes[0] / 2;                   // pos2d is (N,2)
    const int NB  = (N + CHUNK - 1) / CHUNK;           // chunks
    const int CPS = (NB + NSPAN - 1) / NSPAN;          // chunks per span
    int* chunkHist = (int*)d_ws;                       // NB   * 256 ints
    int* spanSum   = chunkHist + (size_t)NB * MAX_BINS; // NSPAN* 256 ints

    gs_hist_kernel<<<NB, 256, 0, stream>>>(pos, ntile, chunkHist, N);
    gs_span_scan_kernel<<<NSPAN, 256, 0, stream>>>(chunkHist, spanSum, NB, CPS);
    gs_bin_scan_kernel<<<1, 256, 0, stream>>>(spanSum, ntile, out, N);
    gs_scatter_kernel<<<NB, 256, 0, stream>>>(pos, cov, opac, ntile, chunkHist,
                                              spanSum, out, N, CPS);
}